// HeteroGNN_68642167324728
// MI455X (gfx1250) — compile-verified
//
#include <hip/hip_runtime.h>
#include <hip/hip_bf16.h>

// ---------------------------------------------------------------------------
// HeteroGNN for MI455X (gfx1250, wave32).
// GEMMs run on V_WMMA_F32_16X16X4_F32 (fp32 matrix pipe, exact numerics).
// Each wave computes a 16x64 output strip: 1 A-load + 4 B-loads -> 4 WMMAs
// per K-step, with weights pre-transposed so all fragment loads are b64.
// Aggregation = edge-parallel global_atomic_add_f32 into L2-resident buffers.
// ---------------------------------------------------------------------------

#define N_G   20000
#define N_D   5000
#define F_IN  256
#define HDIM  128
#define OUTD  64
#define E_DD  160000
#define E_GG  640000
#define E_GD  200000
#define SLOPE 0.01f

typedef __attribute__((ext_vector_type(2))) float v2f;
typedef __attribute__((ext_vector_type(8))) float v8f;

#define WMMA_F32(a, b, c)                                                     \
  __builtin_amdgcn_wmma_f32_16x16x4_f32(false, (a), false, (b), (short)0, (c), \
                                        false, false)

// ---------------- weight transpose: Wt[Hd,K] = W[K,Hd]^T --------------------
__global__ void transpose_kernel(const float* __restrict__ W,
                                 float* __restrict__ Wt, int K, int Hd) {
  int i = blockIdx.x * blockDim.x + threadIdx.x;
  if (i >= K * Hd) return;
  int k = i / Hd, n = i - k * Hd;
  Wt[(size_t)n * K + k] = W[i];
}

// ---------------- WMMA GEMM: Y[N,Hd] = X[N,K] @ W[K,Hd] (+bias) -------------
// Wt is the transposed weight [Hd,K]. Hd must be a multiple of 64.
// One wave per 16-row x 64-col strip; K stepped by 4.
__global__ __launch_bounds__(128) void gemm_wmma_f32(
    const float* __restrict__ X, const float* __restrict__ Wt,
    const float* __restrict__ bias, float* __restrict__ Y,
    int N, int K, int Hd) {
  int wave = (blockIdx.x * blockDim.x + threadIdx.x) >> 5;
  int lane = threadIdx.x & 31;
  int grpH = Hd >> 6;                              // 64-column groups
  int tilesN = (N + 15) >> 4;
  if (wave >= tilesN * grpH) return;
  int tm = wave / grpH, tg = wave - tm * grpH;
  int m0 = tm << 4, n0 = tg << 6;

  // A-frag (ISA 16x4 f32 layout): lane%16 = M row; lane/16 selects K {0,2};
  // VGPR pair holds K,K+1  -> one contiguous b64 per step.
  int mRow   = m0 + (lane & 15);
  int mClamp = mRow < N ? mRow : N - 1;            // tail-tile clamp (loads only)
  int kHalf  = (lane >> 4) << 1;                   // 0 or 2
  int nIn    = lane & 15;

  const float* xr = X + (size_t)mClamp * K + kHalf;
  const float* w0 = Wt + (size_t)(n0 + nIn) * K + kHalf;   // col sub-tile 0
  const size_t wsub = (size_t)16 * K;                      // next 16-col sub-tile

  v8f c0 = {0.f, 0.f, 0.f, 0.f, 0.f, 0.f, 0.f, 0.f};
  v8f c1 = c0, c2 = c0, c3 = c0;
  for (int k = 0; k < K; k += 4) {
    v2f a  = *(const v2f*)xr;
    v2f b0 = *(const v2f*)(w0);
    v2f b1 = *(const v2f*)(w0 + wsub);
    v2f b2 = *(const v2f*)(w0 + 2 * wsub);
    v2f b3 = *(const v2f*)(w0 + 3 * wsub);
    c0 = WMMA_F32(a, b0, c0);
    c1 = WMMA_F32(a, b1, c1);
    c2 = WMMA_F32(a, b2, c2);
    c3 = WMMA_F32(a, b3, c3);
    xr += 4;
    w0 += 4;
  }

  // C/D layout: VGPR r holds M = r + 8*(lane>>4), N = lane%16.
  int rBase = m0 + ((lane >> 4) << 3);
  v8f cc[4] = {c0, c1, c2, c3};
  #pragma unroll
  for (int g = 0; g < 4; ++g) {
    int nCol = n0 + g * 16 + nIn;
    float bv = bias ? bias[nCol] : 0.f;
    #pragma unroll
    for (int r = 0; r < 8; ++r) {
      int row = rBase + r;
      if (row < N) Y[(size_t)row * Hd + nCol] = cc[g][r] + bv;
    }
  }
}

// ---------------- scatter-add aggregation over edges ------------------------
__global__ void aggregate_kernel(const float* __restrict__ Y,
                                 const int* __restrict__ src,
                                 const int* __restrict__ dst,
                                 float* __restrict__ acc,
                                 float* __restrict__ cnt,
                                 int E, int Hd) {
  int chunks = Hd >> 2;                            // float4 chunks per edge
  int idx = blockIdx.x * blockDim.x + threadIdx.x;
  if (idx >= E * chunks) return;
  int e  = idx / chunks;
  int ch = idx - e * chunks;
  int s = src[e], d = dst[e];
  float4 v = *(const float4*)(Y + (size_t)s * Hd + ch * 4);
  float* ap = acc + (size_t)d * Hd + ch * 4;
  __hip_atomic_fetch_add(ap + 0, v.x, __ATOMIC_RELAXED, __HIP_MEMORY_SCOPE_AGENT);
  __hip_atomic_fetch_add(ap + 1, v.y, __ATOMIC_RELAXED, __HIP_MEMORY_SCOPE_AGENT);
  __hip_atomic_fetch_add(ap + 2, v.z, __ATOMIC_RELAXED, __HIP_MEMORY_SCOPE_AGENT);
  __hip_atomic_fetch_add(ap + 3, v.w, __ATOMIC_RELAXED, __HIP_MEMORY_SCOPE_AGENT);
  if (ch == 0)
    __hip_atomic_fetch_add(cnt + d, 1.0f, __ATOMIC_RELAXED, __HIP_MEMORY_SCOPE_AGENT);
}

// ---------------- fused mean + bias + root + leaky --------------------------
__global__ void combine1_kernel(const float* __restrict__ acc,
                                const float* __restrict__ cnt,
                                const float* __restrict__ bias,
                                const float* __restrict__ yr,
                                float* __restrict__ xout, int N, int Hd) {
  int i = blockIdx.x * blockDim.x + threadIdx.x;
  if (i >= N * Hd) return;
  int n = i / Hd, h = i - n * Hd;
  float c = cnt[n]; c = c < 1.f ? 1.f : c;
  float v = acc[i] / c + bias[h] + yr[i];
  xout[i] = v >= 0.f ? v : SLOPE * v;
}

__global__ void combine2_kernel(const float* __restrict__ accA,
                                const float* __restrict__ cntA,
                                const float* __restrict__ biasA,
                                const float* __restrict__ yrA,
                                const float* __restrict__ accB,
                                const float* __restrict__ cntB,
                                const float* __restrict__ biasB,
                                const float* __restrict__ yrB,
                                float* __restrict__ xout, int N, int Hd) {
  int i = blockIdx.x * blockDim.x + threadIdx.x;
  if (i >= N * Hd) return;
  int n = i / Hd, h = i - n * Hd;
  float ca = cntA[n]; ca = ca < 1.f ? 1.f : ca;
  float cb = cntB[n]; cb = cb < 1.f ? 1.f : cb;
  float v = accA[i] / ca + biasA[h] + yrA[i]
          + accB[i] / cb + biasB[h] + yrB[i];
  xout[i] = v >= 0.f ? v : SLOPE * v;
}

// ---------------- misc helpers ----------------------------------------------
__global__ void zero_u32_kernel(unsigned int* p, int n) {
  int i = blockIdx.x * blockDim.x + threadIdx.x;
  if (i < n) p[i] = 0u;
}

__global__ void count_kernel(const int* __restrict__ src, int* cnt, int E) {
  int i = blockIdx.x * blockDim.x + threadIdx.x;
  if (i < E) atomicAdd(&cnt[src[i]], 1);
}

__global__ void prefix_kernel(const int* __restrict__ cnt, int* off, int n) {
  // single-thread serial exclusive scan (20000 elems; negligible vs edge work)
  int s = 0;
  for (int i = 0; i < n; ++i) { off[i] = s; s += cnt[i]; }
  off[n] = s;
}

__global__ void fill_kernel(const int* __restrict__ src, const int* __restrict__ dst,
                            const int* __restrict__ off, int* cur,
                            int* slotSrc, int* slotDst, int E) {
  int i = blockIdx.x * blockDim.x + threadIdx.x;
  if (i >= E) return;
  int s = src[i];
  int pos = off[s] + atomicAdd(&cur[s], 1);
  slotSrc[pos] = s;
  slotDst[pos] = dst[i];
}

__global__ void sort_buckets_kernel(int* slotDst, const int* __restrict__ off, int n) {
  int i = blockIdx.x * blockDim.x + threadIdx.x;
  if (i >= n) return;
  int lo = off[i], hi = off[i + 1];
  for (int a = lo + 1; a < hi; ++a) {       // avg bucket ~10: insertion sort
    int key = slotDst[a];
    int b = a - 1;
    while (b >= lo && slotDst[b] > key) { slotDst[b + 1] = slotDst[b]; --b; }
    slotDst[b + 1] = key;
  }
}

__global__ void score_kernel(const float* __restrict__ G, const float* __restrict__ D,
                             const int* __restrict__ slotSrc,
                             const int* __restrict__ slotDst,
                             float* __restrict__ out, int E) {
  int i = blockIdx.x * blockDim.x + threadIdx.x;
  if (i >= E) return;
  const float4* g = (const float4*)(G + (size_t)slotSrc[i] * OUTD);
  const float4* d = (const float4*)(D + (size_t)slotDst[i] * OUTD);
  float s = 0.f;
  #pragma unroll
  for (int k = 0; k < OUTD / 4; ++k) {
    float4 a = g[k], b = d[k];
    s += a.x * b.x + a.y * b.y + a.z * b.z + a.w * b.w;
  }
  out[i] = s;
}

// ---------------------------------------------------------------------------
extern "C" void kernel_launch(void* const* d_in, const int* in_sizes, int n_in,
                              void* d_out, int out_size, void* d_ws, size_t ws_size,
                              hipStream_t stream) {
  // Input order: top-level dict insertion order; nested params flattened via
  // jax tree_flatten (sorted dict keys: Dlin < Glin < layers; DD < GD < GG).
  const float* xG    = (const float*)d_in[0];
  const float* xD    = (const float*)d_in[1];
  const float* DlinW = (const float*)d_in[2];
  const float* Dlinb = (const float*)d_in[3];
  const float* GlinW = (const float*)d_in[4];
  const float* Glinb = (const float*)d_in[5];
  // layers: base 6 (L0), 15 (L1); per layer: DD(Wl,bl,Wr), GD(Wl,bl,Wr), GG(Wl,bl,Wr)
  const int* eDD     = (const int*)d_in[24];   // [2, E_DD]
  const int* eGG     = (const int*)d_in[25];   // [2, E_GG]
  const int* eGDs    = (const int*)d_in[26];
  const int* eGDd    = (const int*)d_in[27];
  float* out = (float*)d_out;

  // -------- workspace layout (256B aligned slices) --------
  char* base = (char*)d_ws;
  size_t cur = 0;
  auto alloc = [&](size_t bytes) -> void* {
    void* p = base + cur;
    cur = (cur + bytes + 255) & ~(size_t)255;
    return p;
  };
  float* xg1   = (float*)alloc((size_t)N_G * HDIM * 4);
  float* xg2   = (float*)alloc((size_t)N_G * HDIM * 4);
  float* xd1   = (float*)alloc((size_t)N_D * HDIM * 4);
  float* xd2   = (float*)alloc((size_t)N_D * HDIM * 4);
  float* yl    = (float*)alloc((size_t)N_G * HDIM * 4);   // reused per edge type
  float* yrG   = (float*)alloc((size_t)N_G * HDIM * 4);
  float* yrDD  = (float*)alloc((size_t)N_D * HDIM * 4);
  float* yrGD  = (float*)alloc((size_t)N_D * HDIM * 4);
  float* accG  = (float*)alloc((size_t)N_G * HDIM * 4);
  float* cntG  = (float*)alloc((size_t)N_G * 4);
  float* accDD = (float*)alloc((size_t)N_D * HDIM * 4);
  float* cntDD = (float*)alloc((size_t)N_D * 4);
  float* accGD = (float*)alloc((size_t)N_D * HDIM * 4);
  float* cntGD = (float*)alloc((size_t)N_D * 4);
  float* Gfin  = (float*)alloc((size_t)N_G * OUTD * 4);
  float* Dfin  = (float*)alloc((size_t)N_D * OUTD * 4);
  float* wt    = (float*)alloc((size_t)F_IN * HDIM * 4);  // transposed weights
  int* offB    = (int*)alloc((size_t)(N_G + 1) * 4);
  int* curB    = (int*)alloc((size_t)N_G * 4);
  int* slotSrc = (int*)alloc((size_t)E_GD * 4);
  int* slotDst = (int*)alloc((size_t)E_GD * 4);

  auto zero = [&](void* p, size_t words) {
    int n = (int)words;
    zero_u32_kernel<<<(n + 255) / 256, 256, 0, stream>>>((unsigned int*)p, n);
  };
  auto gemm = [&](const float* X, const float* W, const float* bias, float* Y,
                  int N, int K, int Hd) {
    int kh = K * Hd;
    transpose_kernel<<<(kh + 255) / 256, 256, 0, stream>>>(W, wt, K, Hd);
    int waves = ((N + 15) / 16) * (Hd / 64);
    gemm_wmma_f32<<<(waves + 3) / 4, 128, 0, stream>>>(X, wt, bias, Y, N, K, Hd);
  };
  auto aggregate = [&](const float* Y, const int* src, const int* dst,
                       float* acc, float* cnt, int E) {
    int total = E * (HDIM / 4);
    aggregate_kernel<<<(total + 255) / 256, 256, 0, stream>>>(Y, src, dst, acc, cnt,
                                                              E, HDIM);
  };

  const float* xg_in = xG;
  const float* xd_in = xD;
  int Fl = F_IN;
  float* xg_outs[2] = {xg1, xg2};
  float* xd_outs[2] = {xd1, xd2};

  for (int l = 0; l < 2; ++l) {
    int pb = 6 + l * 9;
    const float* WlDD = (const float*)d_in[pb + 0];
    const float* blDD = (const float*)d_in[pb + 1];
    const float* WrDD = (const float*)d_in[pb + 2];
    const float* WlGD = (const float*)d_in[pb + 3];
    const float* blGD = (const float*)d_in[pb + 4];
    const float* WrGD = (const float*)d_in[pb + 5];
    const float* WlGG = (const float*)d_in[pb + 6];
    const float* blGG = (const float*)d_in[pb + 7];
    const float* WrGG = (const float*)d_in[pb + 8];

    zero(accG, (size_t)N_G * HDIM);  zero(cntG, N_G);
    zero(accDD, (size_t)N_D * HDIM); zero(cntDD, N_D);
    zero(accGD, (size_t)N_D * HDIM); zero(cntGD, N_D);

    // GG: G -> G   (transform-first: aggregate 128-dim instead of Fl-dim)
    gemm(xg_in, WlGG, nullptr, yl, N_G, Fl, HDIM);
    aggregate(yl, eGG, eGG + E_GG, accG, cntG, E_GG);
    gemm(xg_in, WrGG, nullptr, yrG, N_G, Fl, HDIM);
    // DD: D -> D
    gemm(xd_in, WlDD, nullptr, yl, N_D, Fl, HDIM);
    aggregate(yl, eDD, eDD + E_DD, accDD, cntDD, E_DD);
    gemm(xd_in, WrDD, nullptr, yrDD, N_D, Fl, HDIM);
    // GD: G -> D
    gemm(xg_in, WlGD, nullptr, yl, N_G, Fl, HDIM);
    aggregate(yl, eGDs, eGDd, accGD, cntGD, E_GD);
    gemm(xd_in, WrGD, nullptr, yrGD, N_D, Fl, HDIM);

    combine1_kernel<<<(N_G * HDIM + 255) / 256, 256, 0, stream>>>(
        accG, cntG, blGG, yrG, xg_outs[l], N_G, HDIM);
    combine2_kernel<<<(N_D * HDIM + 255) / 256, 256, 0, stream>>>(
        accDD, cntDD, blDD, yrDD, accGD, cntGD, blGD, yrGD, xd_outs[l], N_D, HDIM);

    xg_in = xg_outs[l];
    xd_in = xd_outs[l];
    Fl = HDIM;
  }

  // Final linears (WMMA, fused bias)
  gemm(xg_in, GlinW, Glinb, Gfin, N_G, HDIM, OUTD);
  gemm(xd_in, DlinW, Dlinb, Dfin, N_D, HDIM, OUTD);

  // Edge scores in lexicographic (src, dst) order: counting sort by src bucket,
  // then per-bucket insertion sort by dst. Duplicate (src,dst) keys produce
  // identical score values, so output is deterministic despite atomics.
  zero(curB, N_G);
  count_kernel<<<(E_GD + 255) / 256, 256, 0, stream>>>(eGDs, curB, E_GD);
  prefix_kernel<<<1, 1, 0, stream>>>(curB, offB, N_G);
  zero(curB, N_G);
  fill_kernel<<<(E_GD + 255) / 256, 256, 0, stream>>>(eGDs, eGDd, offB, curB,
                                                      slotSrc, slotDst, E_GD);
  sort_buckets_kernel<<<(N_G + 255) / 256, 256, 0, stream>>>(slotDst, offB, N_G);
  score_kernel<<<(E_GD + 255) / 256, 256, 0, stream>>>(Gfin, Dfin, slotSrc,
                                                       slotDst, out, E_GD);
}